// Spatial_Attention_1297080123985
// MI455X (gfx1250) — compile-verified
//
#include <hip/hip_runtime.h>
#include <hip/hip_bf16.h>
#include <limits.h>

typedef float v2f __attribute__((ext_vector_type(2)));
typedef float v8f __attribute__((ext_vector_type(8)));

#define LRELU(x) ((x) > 0.0f ? (x) : 0.2f * (x))

__device__ __forceinline__ v8f wmma4(v2f a, v2f b, v8f c) {
  // D = A(16x4,f32) * B(4x16,f32) + C(16x16,f32)
  return __builtin_amdgcn_wmma_f32_16x16x4_f32(false, a, false, b, (short)0, c,
                                               false, false);
}

// Monotone float<->int encoding so atomicMax(int) implements float segment-max.
__device__ __forceinline__ int f2ord(float f) {
  int i = __float_as_int(f);
  return i >= 0 ? i : (i ^ 0x7fffffff);
}
__device__ __forceinline__ float ord2f(int s) {
  int i = s >= 0 ? s : (s ^ 0x7fffffff);
  return __int_as_float(i);
}

// ---------------------------------------------------------------------------
// Repack W[K,Nc] into WMMA-B fragment order: for k-quad q, half h, col n the
// pair (W[4q+2h][n], W[4q+2h+1][n]) is stored contiguously -> per-lane b64.
// ---------------------------------------------------------------------------
__global__ void packW_kernel(const float* __restrict__ W,
                             float* __restrict__ Wp, int K, int Nc) {
  int i = blockIdx.x * blockDim.x + threadIdx.x;
  if (i >= K * Nc) return;
  int k = i / Nc, n = i - k * Nc;
  int quad = k >> 2, kin = k & 3, half = kin >> 1, pair = kin & 1;
  Wp[(((size_t)(quad * 2 + half)) * Nc + n) * 2 + pair] = W[i];
}

// ---------------------------------------------------------------------------
// f32 WMMA GEMM: C[M,Nc] = A[M,K] @ W[K,Nc] (+bias), W pre-packed (packW).
// block = 128 threads (4 waves). Each wave: 32 rows x 64 cols -> block 128x64.
// grid = (M/128, Nc/64). Per k-step: 8 WMMA vs 6 b64 loads.
// ---------------------------------------------------------------------------
__global__ void gemm_wmma_kernel(const float* __restrict__ A,
                                 const float* __restrict__ Wp,
                                 const float* __restrict__ bias,
                                 float* __restrict__ C, int K, int Nc) {
  const int wave = threadIdx.x >> 5;
  const int lane = threadIdx.x & 31;
  const int half = lane >> 4;
  const int idx = lane & 15;
  const int tileM = blockIdx.x * 128 + wave * 32;
  const int tileN = blockIdx.y * 64;
  const float* __restrict__ arow0 = A + (size_t)(tileM + idx) * K;
  const float* __restrict__ arow1 = A + (size_t)(tileM + 16 + idx) * K;
  const v2f* __restrict__ Wp2 = (const v2f*)Wp;

  v8f acc00 = {}, acc01 = {}, acc02 = {}, acc03 = {};
  v8f acc10 = {}, acc11 = {}, acc12 = {}, acc13 = {};
  for (int k0 = 0; k0 < K; k0 += 4) {
    const int ka = k0 + 2 * half;
    v2f a0 = *(const v2f*)(arow0 + ka);
    v2f a1 = *(const v2f*)(arow1 + ka);
    const v2f* wq = Wp2 + (size_t)((k0 >> 2) * 2 + half) * Nc + tileN + idx;
    v2f b0 = wq[0];
    v2f b1 = wq[16];
    v2f b2 = wq[32];
    v2f b3 = wq[48];
    acc00 = wmma4(a0, b0, acc00);
    acc10 = wmma4(a1, b0, acc10);
    acc01 = wmma4(a0, b1, acc01);
    acc11 = wmma4(a1, b1, acc11);
    acc02 = wmma4(a0, b2, acc02);
    acc12 = wmma4(a1, b2, acc12);
    acc03 = wmma4(a0, b3, acc03);
    acc13 = wmma4(a1, b3, acc13);
  }
  v8f acc0[4] = {acc00, acc01, acc02, acc03};
  v8f acc1[4] = {acc10, acc11, acc12, acc13};
#pragma unroll
  for (int j = 0; j < 4; ++j) {
    const int n = tileN + 16 * j + idx;
    const float bv = bias ? bias[n] : 0.0f;
#pragma unroll
    for (int v = 0; v < 8; ++v) {
      const int mm = tileM + v + 8 * half;
      C[(size_t)mm * Nc + n] = acc0[j][v] + bv;
      C[(size_t)(mm + 16) * Nc + n] = acc1[j][v] + bv;
    }
  }
}

// ---------------------------------------------------------------------------
// Layer-1 attention logits: alpha[e,h] = sum_c lrelu(xl[src]+xr[dst]+e1)*att
// plus segment-max into amax via ordered-int atomicMax. D=256, C=64, H=4.
// ---------------------------------------------------------------------------
__global__ void alpha1_kernel(const float* __restrict__ xl,
                              const float* __restrict__ xr,
                              const float* __restrict__ ebuf,
                              const int* __restrict__ adj,
                              const float* __restrict__ att,
                              float* __restrict__ alpha, int* __restrict__ amax,
                              long long E, int D, int C) {
  long long idx = (long long)blockIdx.x * blockDim.x + threadIdx.x;
  if (idx >= E * 4) return;
  long long e = idx >> 2;
  int h = (int)(idx & 3);
  int src = adj[e];
  int dst = adj[E + e];
  const float4* xls = (const float4*)(xl + (size_t)src * D + h * C);
  const float4* xrd = (const float4*)(xr + (size_t)dst * D + h * C);
  const float4* ep = (const float4*)(ebuf + (size_t)e * D + h * C);
  const float4* ah = (const float4*)(att + h * C);
  float s = 0.0f;
  for (int c = 0; c < C / 4; ++c) {
    float4 a = xls[c], b = xrd[c], g = ep[c], w = ah[c];
    float m0 = LRELU(a.x + b.x + g.x);
    float m1 = LRELU(a.y + b.y + g.y);
    float m2 = LRELU(a.z + b.z + g.z);
    float m3 = LRELU(a.w + b.w + g.w);
    s += m0 * w.x + m1 * w.y + m2 * w.z + m3 * w.w;
  }
  alpha[idx] = s;
  atomicMax(amax + (size_t)dst * 4 + h, f2ord(s));
}

// ---------------------------------------------------------------------------
// Fused layer-2: gather nodeOut rows (edge_efea), WMMA gather-GEMM -> LDS
// (32x128 tile), then alpha2 + segment max. block = 128 thr, 32 edges/block.
// ---------------------------------------------------------------------------
__global__ void fused_alpha2_kernel(
    const float* __restrict__ nodeOut, const float* __restrict__ WeP,
    const int* __restrict__ efea, const int* __restrict__ adj,
    const float* __restrict__ xl, const float* __restrict__ xr,
    const float* __restrict__ att, float* __restrict__ alpha,
    int* __restrict__ amax, long long E2) {
  __shared__ float eT[32 * 128];
  const int wave = threadIdx.x >> 5;
  const int lane = threadIdx.x & 31;
  const int half = lane >> 4;
  const int idx = lane & 15;
  const long long ebase = (long long)blockIdx.x * 32;
  const int grow0 = efea[ebase + idx];
  const int grow1 = efea[ebase + 16 + idx];
  const float* __restrict__ arow0 = nodeOut + (size_t)grow0 * 256;
  const float* __restrict__ arow1 = nodeOut + (size_t)grow1 * 256;
  const v2f* __restrict__ Wp2 = (const v2f*)WeP;
  const int nbase = wave * 32;

  v8f acc00 = {}, acc01 = {}, acc10 = {}, acc11 = {};
  for (int k0 = 0; k0 < 256; k0 += 4) {
    const int ka = k0 + 2 * half;
    v2f a0 = *(const v2f*)(arow0 + ka);
    v2f a1 = *(const v2f*)(arow1 + ka);
    const v2f* wq = Wp2 + (size_t)((k0 >> 2) * 2 + half) * 128 + nbase + idx;
    v2f b0 = wq[0];
    v2f b1 = wq[16];
    acc00 = wmma4(a0, b0, acc00);
    acc10 = wmma4(a1, b0, acc10);
    acc01 = wmma4(a0, b1, acc01);
    acc11 = wmma4(a1, b1, acc11);
  }
  v8f acc0[2] = {acc00, acc01};
  v8f acc1[2] = {acc10, acc11};
#pragma unroll
  for (int j = 0; j < 2; ++j) {
    const int n = nbase + 16 * j + idx;
#pragma unroll
    for (int v = 0; v < 8; ++v) {
      eT[(v + 8 * half) * 128 + n] = acc0[j][v];
      eT[(v + 8 * half + 16) * 128 + n] = acc1[j][v];
    }
  }
  __syncthreads();

  const int t = threadIdx.x;  // 32 edges x 4 heads = 128 pairs
  const int r = t >> 2, h = t & 3;
  const long long e = ebase + r;
  const int src = adj[e];
  const int dst = adj[E2 + e];
  const float4* xls = (const float4*)(xl + (size_t)src * 128 + h * 32);
  const float4* xrd = (const float4*)(xr + (size_t)dst * 128 + h * 32);
  const float4* ep = (const float4*)(eT + r * 128 + h * 32);
  const float4* ah = (const float4*)(att + h * 32);
  float s = 0.0f;
  for (int c = 0; c < 8; ++c) {
    float4 a = xls[c], b = xrd[c], g = ep[c], w = ah[c];
    float m0 = LRELU(a.x + b.x + g.x);
    float m1 = LRELU(a.y + b.y + g.y);
    float m2 = LRELU(a.z + b.z + g.z);
    float m3 = LRELU(a.w + b.w + g.w);
    s += m0 * w.x + m1 * w.y + m2 * w.z + m3 * w.w;
  }
  alpha[e * 4 + h] = s;
  atomicMax(amax + (size_t)dst * 4 + h, f2ord(s));
}

__global__ void expsum_kernel(float* __restrict__ alpha,
                              const int* __restrict__ dstArr,
                              const int* __restrict__ amax,
                              float* __restrict__ denom, long long E) {
  long long idx = (long long)blockIdx.x * blockDim.x + threadIdx.x;
  if (idx >= E * 4) return;
  long long e = idx >> 2;
  int h = (int)(idx & 3);
  int d = dstArr[e];
  float ex = expf(alpha[idx] - ord2f(amax[(size_t)d * 4 + h]));
  alpha[idx] = ex;
  unsafeAtomicAdd(denom + (size_t)d * 4 + h, ex);
}

__global__ void aggregate_kernel(const float* __restrict__ alphaEx,
                                 const float* __restrict__ denom,
                                 const float* __restrict__ xl,
                                 const int* __restrict__ adj,
                                 float* __restrict__ out, long long E, int D,
                                 int C) {
  const int gpr = D >> 2;
  long long idx = (long long)blockIdx.x * blockDim.x + threadIdx.x;
  long long total = E * gpr;
  if (idx >= total) return;
  long long e = idx / gpr;
  int g = (int)(idx - e * gpr);
  int co = g * 4;
  int h = co / C;
  int src = adj[e];
  int dst = adj[E + e];
  float w = alphaEx[e * 4 + h] / (denom[(size_t)dst * 4 + h] + 1e-16f);
  float4 xv = *(const float4*)(xl + (size_t)src * D + co);
  float* op = out + (size_t)dst * D + co;
  unsafeAtomicAdd(op + 0, w * xv.x);
  unsafeAtomicAdd(op + 1, w * xv.y);
  unsafeAtomicAdd(op + 2, w * xv.z);
  unsafeAtomicAdd(op + 3, w * xv.w);
}

__global__ void bias_kernel(float* __restrict__ out,
                            const float* __restrict__ bias, long long total,
                            int D) {
  long long idx = (long long)blockIdx.x * blockDim.x + threadIdx.x;
  if (idx >= total) return;
  out[idx] += bias[(int)(idx & (D - 1))];  // D is power of two
}

__global__ void fillf_kernel(float* __restrict__ p, float v, long long n) {
  long long idx = (long long)blockIdx.x * blockDim.x + threadIdx.x;
  if (idx < n) p[idx] = v;
}
__global__ void filli_kernel(int* __restrict__ p, int v, long long n) {
  long long idx = (long long)blockIdx.x * blockDim.x + threadIdx.x;
  if (idx < n) p[idx] = v;
}

__global__ void concat_kernel(const float* __restrict__ xn,
                              const float* __restrict__ xlg,
                              float* __restrict__ nodeF, int total) {
  int i = blockIdx.x * blockDim.x + threadIdx.x;
  if (i >= total) return;
  int r = i >> 7, c = i & 127;
  nodeF[(size_t)r * 256 + c] = xn[i];
  nodeF[(size_t)r * 256 + 128 + c] = xlg[i];
}

__global__ void split_kernel(const float* __restrict__ nodeOut,
                             float* __restrict__ xn, float* __restrict__ xlg,
                             int total) {
  int i = blockIdx.x * blockDim.x + threadIdx.x;
  if (i >= total) return;
  int r = i >> 7, c = i & 127;
  xn[i] = nodeOut[(size_t)r * 256 + c];
  xlg[i] = nodeOut[(size_t)r * 256 + 128 + c];
}

// ---------------------------------------------------------------------------
extern "C" void kernel_launch(void* const* d_in, const int* in_sizes, int n_in,
                              void* d_out, int out_size, void* d_ws,
                              size_t ws_size, hipStream_t stream) {
  const float* x_node = (const float*)d_in[0];   // [8192,128]
  const float* x_trace = (const float*)d_in[1];  // [131072,128]
  const float* x_log = (const float*)d_in[2];    // [8192,128]
  const int* node_adj = (const int*)d_in[3];     // [2,131072]
  const int* edge_adj = (const int*)d_in[4];     // [2,524288]
  const int* edge_efea = (const int*)d_in[5];    // [524288]
  const float* n2n_Wl = (const float*)d_in[6];
  const float* n2n_bl = (const float*)d_in[7];
  const float* n2n_Wr = (const float*)d_in[8];
  const float* n2n_br = (const float*)d_in[9];
  const float* n2n_We = (const float*)d_in[10];
  const float* n2n_att = (const float*)d_in[11];
  const float* n2n_bias = (const float*)d_in[12];
  const float* e2n_Wl = (const float*)d_in[13];
  const float* e2n_bl = (const float*)d_in[14];
  const float* e2n_Wr = (const float*)d_in[15];
  const float* e2n_br = (const float*)d_in[16];
  const float* e2n_We = (const float*)d_in[17];
  const float* e2n_att = (const float*)d_in[18];
  const float* e2n_bias = (const float*)d_in[19];

  const int N = 8192;
  const long long E = 131072;
  const long long E2 = 524288;

  float* ws = (float*)d_ws;
  float* nodeF = ws;                       // 2,097,152
  float* xl1 = ws + 2097152;               // 2,097,152
  float* xr1 = ws + 4194304;               // 2,097,152
  float* nodeOut = ws + 6291456;           // 2,097,152
  float* e1 = ws + 8388608;                // 33,554,432 (reused below)
  float* xl2 = e1;                         // 16,777,216 (reuse of e1)
  float* xr2 = e1 + 16777216;              // 16,777,216 (reuse of e1)
  float* alpha1 = ws + 41943040;           // 524,288
  int* amax1 = (int*)(ws + 42467328);      // 32,768
  float* denom1 = ws + 42500096;           // 32,768
  float* alpha2 = ws + 42532864;           // 2,097,152
  int* amax2 = (int*)(ws + 44630016);      // 524,288
  float* denom2 = ws + 45154304;           // 524,288
  // packed weights (fragment order)
  float* pWl1 = ws + 45678592;             // 65,536
  float* pWr1 = pWl1 + 65536;              // 65,536
  float* pWe1 = pWr1 + 65536;              // 32,768
  float* pWl2 = pWe1 + 32768;              // 16,384
  float* pWr2 = pWl2 + 16384;              // 16,384
  float* pWe2 = pWr2 + 16384;              // 32,768

  float* outXn = (float*)d_out;            // 1,048,576
  float* outXt = (float*)d_out + 1048576;  // 16,777,216
  float* outXlg = (float*)d_out + 17825792;

  const int T = 256;
  auto blk = [](long long n, int t) { return (unsigned)((n + t - 1) / t); };

  // ---- pack all weight matrices into WMMA-B fragment order ----
  packW_kernel<<<blk(65536, T), T, 0, stream>>>(n2n_Wl, pWl1, 256, 256);
  packW_kernel<<<blk(65536, T), T, 0, stream>>>(n2n_Wr, pWr1, 256, 256);
  packW_kernel<<<blk(32768, T), T, 0, stream>>>(n2n_We, pWe1, 128, 256);
  packW_kernel<<<blk(16384, T), T, 0, stream>>>(e2n_Wl, pWl2, 128, 128);
  packW_kernel<<<blk(16384, T), T, 0, stream>>>(e2n_Wr, pWr2, 128, 128);
  packW_kernel<<<blk(32768, T), T, 0, stream>>>(e2n_We, pWe2, 256, 128);

  // ---- layer 1: node2node GATv2 (D=256, H=4, C=64) ----
  concat_kernel<<<blk(1048576, T), T, 0, stream>>>(x_node, x_log, nodeF,
                                                   1048576);
  gemm_wmma_kernel<<<dim3(N / 128, 4), 128, 0, stream>>>(nodeF, pWl1, n2n_bl,
                                                         xl1, 256, 256);
  gemm_wmma_kernel<<<dim3(N / 128, 4), 128, 0, stream>>>(nodeF, pWr1, n2n_br,
                                                         xr1, 256, 256);
  gemm_wmma_kernel<<<dim3((unsigned)(E / 128), 4), 128, 0, stream>>>(
      x_trace, pWe1, nullptr, e1, 128, 256);

  filli_kernel<<<blk(N * 4, T), T, 0, stream>>>(amax1, INT_MIN, N * 4);
  fillf_kernel<<<blk(N * 4, T), T, 0, stream>>>(denom1, 0.0f, N * 4);
  fillf_kernel<<<blk((long long)N * 256, T), T, 0, stream>>>(
      nodeOut, 0.0f, (long long)N * 256);

  alpha1_kernel<<<blk(E * 4, T), T, 0, stream>>>(xl1, xr1, e1, node_adj,
                                                 n2n_att, alpha1, amax1, E, 256,
                                                 64);
  expsum_kernel<<<blk(E * 4, T), T, 0, stream>>>(alpha1, node_adj + E, amax1,
                                                 denom1, E);
  aggregate_kernel<<<blk(E * 64, T), T, 0, stream>>>(alpha1, denom1, xl1,
                                                     node_adj, nodeOut, E, 256,
                                                     64);
  bias_kernel<<<blk((long long)N * 256, T), T, 0, stream>>>(
      nodeOut, n2n_bias, (long long)N * 256, 256);

  // ---- layer 2: edge2node GATv2 on line graph (D=128, H=4, C=32) ----
  gemm_wmma_kernel<<<dim3((unsigned)(E / 128), 2), 128, 0, stream>>>(
      x_trace, pWl2, e2n_bl, xl2, 128, 128);
  gemm_wmma_kernel<<<dim3((unsigned)(E / 128), 2), 128, 0, stream>>>(
      x_trace, pWr2, e2n_br, xr2, 128, 128);

  filli_kernel<<<blk(E * 4, T), T, 0, stream>>>(amax2, INT_MIN, E * 4);
  fillf_kernel<<<blk(E * 4, T), T, 0, stream>>>(denom2, 0.0f, E * 4);
  fillf_kernel<<<blk(E * 128, T), T, 0, stream>>>(outXt, 0.0f, E * 128);

  // fused gather-GEMM (WMMA) + attention logits + segment max
  fused_alpha2_kernel<<<(unsigned)(E2 / 32), 128, 0, stream>>>(
      nodeOut, pWe2, edge_efea, edge_adj, xl2, xr2, e2n_att, alpha2, amax2, E2);
  expsum_kernel<<<blk(E2 * 4, T), T, 0, stream>>>(alpha2, edge_adj + E2, amax2,
                                                  denom2, E2);
  aggregate_kernel<<<blk(E2 * 32, T), T, 0, stream>>>(alpha2, denom2, xl2,
                                                      edge_adj, outXt, E2, 128,
                                                      32);
  bias_kernel<<<blk(E * 128, T), T, 0, stream>>>(outXt, e2n_bias, E * 128, 128);

  // ---- final output split ----
  split_kernel<<<blk(1048576, T), T, 0, stream>>>(nodeOut, outXn, outXlg,
                                                  1048576);
}